// get_loss_49847390437533
// MI455X (gfx1250) — compile-verified
//
#include <hip/hip_runtime.h>

typedef float v2f __attribute__((ext_vector_type(2)));
typedef float v8f __attribute__((ext_vector_type(8)));

#define B_   4
#define N_   4096
#define C_   100
#define TI_  (N_ / 16)      // 256 tiles per dimension
#define EPS_ 1e-12f

// ---------------------------------------------------------------- zero hist
__global__ void zero_hist_kernel(int* __restrict__ hist) {
    for (int i = threadIdx.x; i < B_ * C_; i += blockDim.x) hist[i] = 0;
}

// ------------------------------------------------- labels (argmax) + histogram
__global__ void labels_kernel(const float* __restrict__ masks,
                              int* __restrict__ labels,
                              int* __restrict__ hist) {
    int idx = blockIdx.x * blockDim.x + threadIdx.x;   // 0 .. B*N-1
    int b = idx / N_;
    int n = idx - b * N_;
    const float* mp = masks + (size_t)b * C_ * N_ + n; // masks[b, c, n], stride N over c
    float best = mp[0];
    int bi = 0;
#pragma unroll 4
    for (int c = 1; c < C_; ++c) {
        float v = mp[(size_t)c * N_];
        if (v > best) { best = v; bi = c; }            // strict > keeps first (argmax tie rule)
    }
    labels[idx] = bi;
    atomicAdd(&hist[b * C_ + bi], 1);                  // integer atomic -> deterministic value
}

// --------------------------------------------- hist -> 1/count (precise, once)
__global__ void inv_hist_kernel(const int* __restrict__ hist,
                                float* __restrict__ invh) {
    int i = blockIdx.x * blockDim.x + threadIdx.x;
    if (i < B_ * C_) {
        int c = hist[i];
        invh[i] = 1.0f / (float)(c > 0 ? c : 1);       // precise div, matches reference rounding
    }
}

// ------------------------------------------------------------- tiled WMMA loss
// One block = one (batch, 16-row band). 8 waves sweep the 256 column tiles.
// Gram tile via V_WMMA_F32_16X16X4_F32 (K=3 padded to 4), fp32 throughout.
// Epilogue is fully branchless: raw v_sqrt_f32 + cndmask selects + FMA accumulate.
__global__ void __launch_bounds__(256) loss_kernel(const float* __restrict__ xyz,
                                                   const int*   __restrict__ labels,
                                                   const float* __restrict__ invh,
                                                   float*       __restrict__ partials) {
    __shared__ float s_sqp[16], s_sqn[16];
    __shared__ int   s_lab[16];
    __shared__ float s_red[256];

    const int b     = blockIdx.x >> 8;        // / TI_
    const int ti    = blockIdx.x & (TI_ - 1);
    const int ibase = ti * 16;
    const int tid   = threadIdx.x;
    const int lane  = tid & 31;
    const int wave  = tid >> 5;
    const int half  = lane >> 4;              // 0: lanes 0-15, 1: lanes 16-31
    const int m     = lane & 15;

    if (tid < 16) {
        const float* xr = xyz + ((size_t)b * N_ + ibase + tid) * 6;
        float x0 = xr[0], x1 = xr[1], x2 = xr[2];
        float x3 = xr[3], x4 = xr[4], x5 = xr[5];
        s_sqp[tid] = x0 * x0 + x1 * x1 + x2 * x2;
        s_sqn[tid] = x3 * x3 + x4 * x4 + x5 * x5;
        s_lab[tid] = labels[b * N_ + ibase + tid];
    }
    __syncthreads();

    // A matrix 16x4 (M x K), K=3 zero-padded to 4.
    // ISA layout: lanes 0-15 hold K=0 (v0), K=1 (v1); lanes 16-31 hold K=2 (v0), K=3 (v1).
    const float* xr = xyz + ((size_t)b * N_ + ibase + m) * 6;
    v2f aP, aN;
    aP[0] = half ? xr[2] : xr[0];
    aP[1] = half ? 0.0f  : xr[1];
    aN[0] = half ? xr[5] : xr[3];
    aN[1] = half ? 0.0f  : xr[4];

    // C/D layout: VGPR r, lane l -> element (M = r + 8*(l>=16), N = l%16)
    float sqp_i[8], sqn_i[8];
    int   lab_i[8];
#pragma unroll
    for (int r = 0; r < 8; ++r) {
        int ri = r + (half << 3);
        sqp_i[r] = s_sqp[ri];
        sqn_i[r] = s_sqn[ri];
        lab_i[r] = s_lab[ri];
    }

    float acc = 0.0f;
    for (int jt = wave; jt < TI_; jt += 8) {   // uniform per wave -> EXEC all-ones at WMMA
        const int jn = jt * 16 + m;
        const float* yr = xyz + ((size_t)b * N_ + jn) * 6;
        float y0 = yr[0], y1 = yr[1], y2 = yr[2];
        float y3 = yr[3], y4 = yr[4], y5 = yr[5];
        float sqp_j = y0 * y0 + y1 * y1 + y2 * y2;
        float sqn_j = y3 * y3 + y4 * y4 + y5 * y5;
        int   lab_j = labels[b * N_ + jn];
        float inv_c = invh[b * C_ + lab_j];

        // B matrix 4x16 (K x N): lane = column; lanes 0-15 hold K=0/1, lanes 16-31 hold K=2/3.
        v2f bP, bN;
        bP[0] = half ? y2 : y0;  bP[1] = half ? 0.0f : y1;
        bN[0] = half ? y5 : y3;  bN[1] = half ? 0.0f : y4;

        v8f c0 = {};
        v8f gP = __builtin_amdgcn_wmma_f32_16x16x4_f32(false, aP, false, bP, (short)0, c0, false, false);
        v8f gN = __builtin_amdgcn_wmma_f32_16x16x4_f32(false, aN, false, bN, (short)0, c0, false, false);

#pragma unroll
        for (int r = 0; r < 8; ++r) {
            float d2p = __builtin_fmaxf(__builtin_fmaf(-2.0f, gP[r], sqp_i[r] + sqp_j), 0.0f);
            float d2n = __builtin_fmaxf(__builtin_fmaf(-2.0f, gN[r], sqn_i[r] + sqn_j), 0.0f);
            float sp  = __builtin_amdgcn_sqrtf(d2p);   // raw v_sqrt_f32 (TRANS, co-executes)
            float sn  = __builtin_amdgcn_sqrtf(d2n);
            float dp  = (d2p > EPS_) ? sp : 0.0f;      // v_cndmask, branchless
            float dn  = (d2n > EPS_) ? sn : 0.0f;
            float w   = (lab_i[r] == lab_j) ? inv_c : 0.0f;
            acc = __builtin_fmaf(__builtin_fmaf(1.5f, dn, dp), w, acc);
        }
    }

    // deterministic block reduction
    s_red[tid] = acc;
    __syncthreads();
    for (int k = 128; k > 0; k >>= 1) {
        if (tid < k) s_red[tid] += s_red[tid + k];
        __syncthreads();
    }
    if (tid == 0) partials[blockIdx.x] = s_red[0];
}

// ------------------------------------------------------------- final reduce
__global__ void reduce_kernel(const float* __restrict__ partials, float* __restrict__ out) {
    __shared__ float red[256];
    int t = threadIdx.x;
    float s = 0.0f;
    for (int i = t; i < B_ * TI_; i += 256) s += partials[i];  // fixed order
    red[t] = s;
    __syncthreads();
    for (int k = 128; k > 0; k >>= 1) {
        if (t < k) red[t] += red[t + k];
        __syncthreads();
    }
    if (t == 0) out[0] = red[0] / (float)(B_ * C_);
}

// ---------------------------------------------------------------- launcher
extern "C" void kernel_launch(void* const* d_in, const int* in_sizes, int n_in,
                              void* d_out, int out_size, void* d_ws, size_t ws_size,
                              hipStream_t stream) {
    (void)in_sizes; (void)n_in; (void)out_size; (void)ws_size;
    const float* xyz   = (const float*)d_in[0];   // (B, N, 6) f32
    const float* masks = (const float*)d_in[1];   // (B, C, N) f32
    float* out = (float*)d_out;                   // scalar f32

    char* ws = (char*)d_ws;
    int*   labels   = (int*)ws;                          // B*N ints      = 65536 B
    int*   hist     = (int*)(ws + 65536);                // B*C ints      =  1600 B (pad 2048)
    float* invh     = (float*)(ws + 65536 + 2048);       // B*C floats    =  1600 B (pad 2048)
    float* partials = (float*)(ws + 65536 + 4096);       // B*TI_ floats  =  4096 B

    zero_hist_kernel<<<1, 128, 0, stream>>>(hist);
    labels_kernel<<<(B_ * N_) / 256, 256, 0, stream>>>(masks, labels, hist);
    inv_hist_kernel<<<(B_ * C_ + 127) / 128, 128, 0, stream>>>(hist, invh);
    loss_kernel<<<B_ * TI_, 256, 0, stream>>>(xyz, labels, invh, partials);
    reduce_kernel<<<1, 256, 0, stream>>>(partials, out);
}